// PoolAggregator_76433238000426
// MI455X (gfx1250) — compile-verified
//
#include <hip/hip_runtime.h>

// ---------------------------------------------------------------------------
// PoolAggregator on MI455X (gfx1250, wave32, WMMA + async-to-LDS)
//   N=50000 nodes, K=25 neighbors, F_IN=128, T=256, OUT=128, R=N*K=1,250,000
//   BN1 stats computed analytically from Gram(X) -> big GEMM runs once, fused
//   with affine + max-pool.  BN2 via stats pass + apply pass of small GEMMs.
//   k_gram double-buffers its LDS staging with GLOBAL_LOAD_ASYNC_TO_LDS_B128
//   (ASYNCcnt-tracked) so HBM streaming overlaps the WMMA K-loop.
// ---------------------------------------------------------------------------

#define NN    50000
#define KK    25
#define FIN   128
#define TT    256
#define OUTF  128
#define RROWS (NN * KK)        // 1,250,000
#define EPS   1e-3f

typedef float v2f __attribute__((ext_vector_type(2)));
typedef float v8f __attribute__((ext_vector_type(8)));

#define WMMA_F32(a, b, c) \
  __builtin_amdgcn_wmma_f32_16x16x4_f32(false, (a), false, (b), (short)0, (c), false, false)

#if __has_builtin(__builtin_amdgcn_global_load_async_to_lds_b128) && \
    __has_builtin(__builtin_amdgcn_s_wait_asynccnt)
#define ASYNC_LDS 1
#else
#define ASYNC_LDS 0
#endif

// ---------------------------------------------------------------------------
// 0) zero the stats region of the workspace (G, s, BN scratch)
// ---------------------------------------------------------------------------
__global__ __launch_bounds__(256) void k_init(float* stats, int n) {
  int i = blockIdx.x * 256 + threadIdx.x;
  if (i < n) stats[i] = 0.0f;
}

// ---------------------------------------------------------------------------
// 1) G = X^T X (128x128) and s = column sums of X, X = neigh_nodes as [R,128]
//    Grid-stride over 40-row chunks (R % 40 == 0 -> no tail masking); 64
//    G-tiles (16x16) held in VGPR accumulators (8 per wave), atomically
//    flushed once at the end.  Double-buffered async global->LDS staging.
// ---------------------------------------------------------------------------
#define CH 40  // rows per chunk; 40*128 floats = 1280 x B128 = 5 per thread

#if ASYNC_LDS
// builtin signature (from hipcc diagnostic): param0 = int4 addrspace(1)*
// (printed as "__device__"), then LDS-side int4 addrspace(3)*, imm offset,
// imm cpol.
typedef int v4i __attribute__((vector_size(16)));
typedef __attribute__((address_space(1))) v4i gv4i_t;
typedef __attribute__((address_space(3))) v4i lv4i_t;

__device__ __forceinline__ void stage_async(const float* gbase, float* lbase) {
  gv4i_t* g = (gv4i_t*)gbase;  // global row pitch: 128 floats = 32 x v4i
  lv4i_t* l = (lv4i_t*)lbase;  // LDS row pitch: 132 floats = 33 x v4i
#pragma unroll
  for (int i = 0; i < 5; ++i) {
    const int idx = threadIdx.x + i * 256;  // B128 transfer index 0..1279
    const int rr  = idx >> 5;               // row 0..39 (32 B128 per row)
    const int cc  = idx & 31;               // B128 column within row
    __builtin_amdgcn_global_load_async_to_lds_b128(g + rr * 32 + cc,
                                                   l + rr * 33 + cc, 0, 0);
  }
}
#endif

__global__ __launch_bounds__(256) void k_gram(const float* __restrict__ X,
                                              float* __restrict__ G,
                                              float* __restrict__ s) {
  __shared__ float lds[2][CH * 132];
  const int tid   = threadIdx.x;
  const int wave  = tid >> 5;
  const int lane  = tid & 31;
  const int lrow  = lane & 15;
  const int khalf = lane >> 4;

  v8f acc[8];
#pragma unroll
  for (int i = 0; i < 8; ++i) acc[i] = (v8f){0, 0, 0, 0, 0, 0, 0, 0};
  float ssum = 0.0f;

  const int nch = RROWS / CH;  // 31250, exact

#if ASYNC_LDS
  int buf = 0;
  const int ch0 = blockIdx.x;
  if (ch0 < nch) stage_async(X + (long)ch0 * CH * FIN, &lds[0][0]);
  for (int ch = ch0; ch < nch; ch += gridDim.x) {
    const int  chn = ch + gridDim.x;
    const bool nxt = chn < nch;
    if (nxt) stage_async(X + (long)chn * CH * FIN, &lds[buf ^ 1][0]);
    // async loads complete in order: <=5 outstanding => this buffer landed
    if (nxt) __builtin_amdgcn_s_wait_asynccnt(5);
    else     __builtin_amdgcn_s_wait_asynccnt(0);
    __syncthreads();
    const float* cur = &lds[buf][0];
#else
  for (int ch = blockIdx.x; ch < nch; ch += gridDim.x) {
    __syncthreads();  // protect LDS from previous iteration's readers
    const long base = (long)ch * CH * FIN;
    for (int idx = tid; idx < CH * FIN; idx += 256) {
      const int rr = idx >> 7, cc = idx & 127;
      lds[0][rr * 132 + cc] = X[base + (long)rr * FIN + cc];
    }
    __syncthreads();
    const float* cur = &lds[0][0];
#endif

    if (tid < 128) {  // column partial sums
      float t = 0.0f;
      for (int rr = 0; rr < CH; ++rr) t += cur[rr * 132 + tid];
      ssum += t;
    }

#pragma unroll
    for (int i = 0; i < 8; ++i) {
      const int tileid = wave * 8 + i;
      const int m0 = (tileid >> 3) * 16, n0 = (tileid & 7) * 16;
      v8f c = acc[i];
      for (int kk = 0; kk < CH / 4; ++kk) {
        const int kr = kk * 4 + 2 * khalf;  // row-of-X == K dim of the GEMM
        v2f a, b;
        a.x = cur[kr * 132 + m0 + lrow];
        a.y = cur[(kr + 1) * 132 + m0 + lrow];
        b.x = cur[kr * 132 + n0 + lrow];
        b.y = cur[(kr + 1) * 132 + n0 + lrow];
        c = WMMA_F32(a, b, c);
      }
      acc[i] = c;
    }

#if ASYNC_LDS
    __syncthreads();  // all waves done with lds[buf] before next issue
    buf ^= 1;
#endif
  }

  if (tid < 128) atomicAdd(&s[tid], ssum);
#pragma unroll
  for (int i = 0; i < 8; ++i) {
    const int tileid = wave * 8 + i;
    const int m0 = (tileid >> 3) * 16, n0 = (tileid & 7) * 16;
#pragma unroll
    for (int r = 0; r < 8; ++r) {
      const int mm = m0 + r + khalf * 8;
      atomicAdd(&G[mm * FIN + n0 + lrow], acc[i][r]);
    }
  }
}

// ---------------------------------------------------------------------------
// 2) BN1 affine params per channel t:  mean/var from (G, s, Wt, bt)
// ---------------------------------------------------------------------------
__global__ __launch_bounds__(256) void k_bn1(const float* __restrict__ G,
                                             const float* __restrict__ s,
                                             const float* __restrict__ Wt,
                                             const float* __restrict__ bt,
                                             const float* __restrict__ g1,
                                             const float* __restrict__ b1,
                                             float* __restrict__ scale1,
                                             float* __restrict__ shift1) {
  const int t = threadIdx.x;  // channel 0..255
  float sw = 0.0f, quad = 0.0f;
  for (int j = 0; j < FIN; ++j) {
    const float wj = Wt[j * TT + t];
    sw += s[j] * wj;
    float gw = 0.0f;
    const float* Gr = &G[j * FIN];
    for (int k = 0; k < FIN; ++k) gw += Gr[k] * Wt[k * TT + t];
    quad += wj * gw;
  }
  const float invR = 1.0f / (float)RROWS;
  const float btt  = bt[t];
  const float mean = sw * invR + btt;
  const float ex2  = quad * invR + 2.0f * btt * sw * invR + btt * btt;
  const float var  = ex2 - mean * mean;
  const float inv  = rsqrtf(var + EPS);
  const float sc   = g1[t] * inv;
  scale1[t] = sc;
  shift1[t] = b1[t] - mean * sc;
}

// ---------------------------------------------------------------------------
// 3) Per node: y = affine(bn1)(neigh_row @ Wt), pooled = max over 25 rows.
//    One workgroup per node: 2 M-tiles (rows 0..31, padded) x 16 N-tiles.
// ---------------------------------------------------------------------------
__global__ __launch_bounds__(256) void k_neigh(const float* __restrict__ X,
                                               const float* __restrict__ Wt,
                                               const float* __restrict__ scale1,
                                               const float* __restrict__ shift1,
                                               float* __restrict__ pooled) {
  __shared__ float ldsA[32 * 132];
  __shared__ float ldsP[2 * TT];
  const int n     = blockIdx.x;
  const int tid   = threadIdx.x;
  const int wave  = tid >> 5;
  const int lane  = tid & 31;
  const int lrow  = lane & 15;
  const int khalf = lane >> 4;

  const long base = (long)n * (KK * FIN);
  for (int idx = tid; idx < 32 * FIN; idx += 256) {
    const int rr = idx >> 7, cc = idx & 127;
    ldsA[rr * 132 + cc] = (rr < KK) ? X[base + idx] : 0.0f;
  }
  __syncthreads();

  const int mtile = wave & 1;
  const int q     = wave >> 1;  // 0..3
#pragma unroll
  for (int i = 0; i < 4; ++i) {
    const int ntile = q + 4 * i;
    const int col   = ntile * 16 + lrow;
    v8f c = (v8f){0, 0, 0, 0, 0, 0, 0, 0};
    for (int kk = 0; kk < FIN / 4; ++kk) {
      const int kr = kk * 4 + 2 * khalf;
      v2f a, b;
      a.x = ldsA[(mtile * 16 + lrow) * 132 + kr];
      a.y = ldsA[(mtile * 16 + lrow) * 132 + kr + 1];
      b.x = Wt[kr * TT + col];
      b.y = Wt[(kr + 1) * TT + col];
      c = WMMA_F32(a, b, c);
    }
    const float sc = scale1[col], sh = shift1[col];
    float m = -3.0e38f;
#pragma unroll
    for (int r = 0; r < 8; ++r) {
      const int mr = mtile * 16 + r + khalf * 8;
      const float y = c[r] * sc + sh;
      if (mr < KK) m = fmaxf(m, y);
    }
    m = fmaxf(m, __shfl_xor(m, 16, 32));  // combine the two half-wave M-halves
    if (lane < 16) ldsP[mtile * TT + col] = m;
  }
  __syncthreads();
  pooled[(long)n * TT + tid] = fmaxf(ldsP[tid], ldsP[TT + tid]);
}

// ---------------------------------------------------------------------------
// 4) out_pre = [self @ Ws | pooled @ Wn] for 16 nodes per workgroup.
//    mode 0: accumulate per-column sum / sumsq (BN2 training stats)
//    mode 1: apply BN2 affine + ReLU, write d_out
// ---------------------------------------------------------------------------
__global__ __launch_bounds__(256) void k_out(const float* __restrict__ selfN,
                                             const float* __restrict__ pooled,
                                             const float* __restrict__ Ws,
                                             const float* __restrict__ Wn,
                                             float* __restrict__ sum2,
                                             float* __restrict__ sumsq2,
                                             const float* __restrict__ scale2,
                                             const float* __restrict__ shift2,
                                             float* __restrict__ outp,
                                             int mode) {
  __shared__ float ldsS[16 * 132];
  __shared__ float ldsPo[16 * 260];
  const int node0 = blockIdx.x * 16;
  const int tid   = threadIdx.x;
  const int wave  = tid >> 5;
  const int lane  = tid & 31;
  const int lrow  = lane & 15;
  const int khalf = lane >> 4;

  for (int idx = tid; idx < 16 * FIN; idx += 256) {
    const int rr = idx >> 7, cc = idx & 127;
    ldsS[rr * 132 + cc] = selfN[(long)(node0 + rr) * FIN + cc];
  }
  for (int idx = tid; idx < 16 * TT; idx += 256) {
    const int rr = idx >> 8, cc = idx & 255;
    ldsPo[rr * 260 + cc] = pooled[(long)(node0 + rr) * TT + cc];
  }
  __syncthreads();

  const int col = wave * 16 + lrow;  // wave w owns output cols [16w,16w+16)
  v8f cs = (v8f){0, 0, 0, 0, 0, 0, 0, 0};
  v8f cp = (v8f){0, 0, 0, 0, 0, 0, 0, 0};
  for (int kk = 0; kk < FIN / 4; ++kk) {  // self @ Ws, K = 128
    const int kr = kk * 4 + 2 * khalf;
    v2f a, b;
    a.x = ldsS[lrow * 132 + kr];
    a.y = ldsS[lrow * 132 + kr + 1];
    b.x = Ws[kr * OUTF + col];
    b.y = Ws[(kr + 1) * OUTF + col];
    cs = WMMA_F32(a, b, cs);
  }
  for (int kk = 0; kk < TT / 4; ++kk) {  // pooled @ Wn, K = 256
    const int kr = kk * 4 + 2 * khalf;
    v2f a, b;
    a.x = ldsPo[lrow * 260 + kr];
    a.y = ldsPo[lrow * 260 + kr + 1];
    b.x = Wn[kr * OUTF + col];
    b.y = Wn[(kr + 1) * OUTF + col];
    cp = WMMA_F32(a, b, cp);
  }

  if (mode == 0) {
    float s1 = 0, q1 = 0, s2 = 0, q2 = 0;
#pragma unroll
    for (int r = 0; r < 8; ++r) {
      const float ya = cs[r], yb = cp[r];
      s1 += ya; q1 += ya * ya;
      s2 += yb; q2 += yb * yb;
    }
    s1 += __shfl_xor(s1, 16, 32); q1 += __shfl_xor(q1, 16, 32);
    s2 += __shfl_xor(s2, 16, 32); q2 += __shfl_xor(q2, 16, 32);
    if (lane < 16) {
      atomicAdd(&sum2[col], s1);          atomicAdd(&sumsq2[col], q1);
      atomicAdd(&sum2[OUTF + col], s2);   atomicAdd(&sumsq2[OUTF + col], q2);
    }
  } else {
    const float scA = scale2[col],        shA = shift2[col];
    const float scB = scale2[OUTF + col], shB = shift2[OUTF + col];
#pragma unroll
    for (int r = 0; r < 8; ++r) {
      const int row = node0 + r + khalf * 8;
      outp[(long)row * (2 * OUTF) + col]        = fmaxf(cs[r] * scA + shA, 0.0f);
      outp[(long)row * (2 * OUTF) + OUTF + col] = fmaxf(cp[r] * scB + shB, 0.0f);
    }
  }
}

// ---------------------------------------------------------------------------
// 5) BN2 affine params from accumulated stats
// ---------------------------------------------------------------------------
__global__ __launch_bounds__(256) void k_bn2(const float* __restrict__ sum2,
                                             const float* __restrict__ sumsq2,
                                             const float* __restrict__ g2,
                                             const float* __restrict__ b2,
                                             float* __restrict__ scale2,
                                             float* __restrict__ shift2) {
  const int t = threadIdx.x;
  const float invN = 1.0f / (float)NN;
  const float mean = sum2[t] * invN;
  const float var  = sumsq2[t] * invN - mean * mean;
  const float inv  = rsqrtf(var + EPS);
  const float sc   = g2[t] * inv;
  scale2[t] = sc;
  shift2[t] = b2[t] - mean * sc;
}

// ---------------------------------------------------------------------------
extern "C" void kernel_launch(void* const* d_in, const int* in_sizes, int n_in,
                              void* d_out, int out_size, void* d_ws, size_t ws_size,
                              hipStream_t stream) {
  const float* selfN = (const float*)d_in[0];   // [N,128]
  const float* neigh = (const float*)d_in[1];   // [N,25,128]
  // d_in[2] = len_adj_nodes (unused by reference)
  const float* Wt = (const float*)d_in[3];      // [128,256]
  const float* bt = (const float*)d_in[4];      // [256]
  const float* g1 = (const float*)d_in[5];      // [256]
  const float* b1 = (const float*)d_in[6];      // [256]
  const float* Wn = (const float*)d_in[7];      // [256,128]
  const float* Ws = (const float*)d_in[8];      // [128,128]
  const float* g2 = (const float*)d_in[9];      // [256]
  const float* b2 = (const float*)d_in[10];     // [256]
  float* outp = (float*)d_out;

  // workspace layout (floats)
  float* G      = (float*)d_ws;          // 128*128
  float* s      = G + FIN * FIN;         // 128
  float* scale1 = s + FIN;               // 256
  float* shift1 = scale1 + TT;           // 256
  float* sum2   = shift1 + TT;           // 256
  float* sumsq2 = sum2 + TT;             // 256
  float* scale2 = sumsq2 + TT;           // 256
  float* shift2 = scale2 + TT;           // 256
  float* pooled = shift2 + TT;           // N*256
  const int statsN = FIN * FIN + FIN + 6 * TT;  // region that must start at 0

  k_init<<<(statsN + 255) / 256, 256, 0, stream>>>(G, statsN);
  k_gram<<<512, 256, 0, stream>>>(neigh, G, s);
  k_bn1<<<1, 256, 0, stream>>>(G, s, Wt, bt, g1, b1, scale1, shift1);
  k_neigh<<<NN, 256, 0, stream>>>(neigh, Wt, scale1, shift1, pooled);
  k_out<<<NN / 16, 256, 0, stream>>>(selfN, pooled, Ws, Wn, sum2, sumsq2,
                                     scale2, shift2, outp, 0);
  k_bn2<<<1, 256, 0, stream>>>(sum2, sumsq2, g2, b2, scale2, shift2);
  k_out<<<NN / 16, 256, 0, stream>>>(selfN, pooled, Ws, Wn, sum2, sumsq2,
                                     scale2, shift2, outp, 1);
}